// MultiHeadAttention_2052994367525
// MI455X (gfx1250) — compile-verified
//
#include <hip/hip_runtime.h>
#include <hip/hip_bf16.h>

typedef float v2f __attribute__((ext_vector_type(2)));
typedef float v8f __attribute__((ext_vector_type(8)));

// ---------------------------------------------------------------------------
// softmax over a size-1 axis == 1.0 exactly, so head_out[h,n,:] == v_proj[h,0,:]
// and q/Wq/Wk/k_param are numerically dead. Surviving work:
//   vc[h*128+k] = sum_d v_param[d] * Wv[h,d,k]     (reads 16 MiB)
//   r[j]        = sum_i vc[i]      * Wo[i,j]       (reads 16 MiB)
//   out[n,:]    = r                                (writes 64 MiB)
// ~96 MiB -> ~4.1 us at 23.3 TB/s. GEMVs run on V_WMMA_F32_16X16X4_F32 (full
// fp32) with the vector in row 0 of the A tile; K is split across 8 waves per
// block with an LDS partial reduction to break the WMMA dependency chain.
// ---------------------------------------------------------------------------

// Block = 256 threads = 8 waves. Each block produces 16 columns of
// out = v (1xK) * B (KxN); wave w handles K slice [w*KSLICE, (w+1)*KSLICE).
//   A 16x4 fp32 tile: lanes 0-15 -> M=lane, K={kk,kk+1}; lanes 16-31 -> K={kk+2,kk+3}
//   B 4x16 fp32 tile: lanes 0-15 -> N=lane, K={kk,kk+1}; lanes 16-31 -> K={kk+2,kk+3}
//   C/D 16x16 fp32:   row M=0 is acc VGPR0 of lanes 0-15.
template <int LDB, int KSLICE>
__global__ void __launch_bounds__(256)
gemv_wmma_f32(const float* __restrict__ v, const float* __restrict__ B,
              float* __restrict__ out, long long strideB, int strideOut)
{
    const int tid  = threadIdx.x;
    const int wave = tid >> 5;
    const int lane = tid & 31;
    const bool hi  = (lane & 16) != 0;   // lanes 16-31 -> K pair {2,3}
    const int  l   = lane & 15;
    const bool row0 = (l == 0);          // only matrix row M==0 carries v
    const int  n0  = blockIdx.x * 16;

    const float* __restrict__ Bh = B + (long long)blockIdx.y * strideB;
    const int k0 = wave * KSLICE;

    // element index of (K row = k0 + 2*hi, column n0+l); LDB is a compile-time
    // constant so the +LDB/+4*LDB/... offsets fold into load immediates.
    unsigned idx = (unsigned)(k0 + (hi ? 2 : 0)) * LDB + (unsigned)(n0 + l);

    v8f acc0 = {}, acc1 = {};
#pragma unroll 4
    for (int kk = 0; kk < KSLICE; kk += 8) {
        // ---- K = k0+kk .. k0+kk+3 ----
        const float4 vv0 = *(const float4*)(v + k0 + kk);   // one b128, wave-broadcast
        v2f a0, b0;
        {
            float t0 = hi ? vv0.z : vv0.x;
            float t1 = hi ? vv0.w : vv0.y;
            a0.x = row0 ? t0 : 0.0f;
            a0.y = row0 ? t1 : 0.0f;
        }
        b0.x = Bh[idx];
        b0.y = Bh[idx + LDB];
        acc0 = __builtin_amdgcn_wmma_f32_16x16x4_f32(
            false, a0, false, b0, (short)0, acc0, false, false);

        // ---- K = k0+kk+4 .. k0+kk+7 ----
        const float4 vv1 = *(const float4*)(v + k0 + kk + 4);
        v2f a1, b1;
        {
            float t0 = hi ? vv1.z : vv1.x;
            float t1 = hi ? vv1.w : vv1.y;
            a1.x = row0 ? t0 : 0.0f;
            a1.y = row0 ? t1 : 0.0f;
        }
        b1.x = Bh[idx + 4 * LDB];
        b1.y = Bh[idx + 5 * LDB];
        acc1 = __builtin_amdgcn_wmma_f32_16x16x4_f32(
            false, a1, false, b1, (short)0, acc1, false, false);

        idx += 8u * LDB;
    }

    // cross-wave reduction of result row M=0 (acc VGPR0, lanes 0-15)
    __shared__ float partial[8][16];
    if (!hi) partial[wave][l] = acc0[0] + acc1[0];
    __syncthreads();
    if (tid < 16) {
        float s = 0.0f;
#pragma unroll
        for (int w = 0; w < 8; ++w) s += partial[w][tid];
        out[(long long)blockIdx.y * strideOut + n0 + tid] = s;
    }
}

// out[row, :] = r  with 128-bit stores; colsVec = cols/4.
__global__ void __launch_bounds__(256)
broadcast_row_f32(const float* __restrict__ r, float* __restrict__ out, int colsVec)
{
    const int cv = blockIdx.x * blockDim.x + threadIdx.x;
    if (cv >= colsVec) return;
    const float4 val = ((const float4*)r)[cv];
    ((float4*)out)[(long long)blockIdx.y * colsVec + cv] = val;
}

extern "C" void kernel_launch(void* const* d_in, const int* in_sizes, int n_in,
                              void* d_out, int out_size, void* d_ws, size_t ws_size,
                              hipStream_t stream)
{
    // setup_inputs order: q, Wq, Wk, Wv, k_param, v_param, Wo  (all fp32)
    const float* Wv = (const float*)d_in[3];   // (16, 2048, 128)
    const float* vp = (const float*)d_in[5];   // (1, 2048)
    const float* Wo = (const float*)d_in[6];   // (2048, 2048)
    float* out = (float*)d_out;                // (8192, 2048) fp32

    float* vc = (float*)d_ws;                  // 2048 floats (concat v_proj)
    float* r  = vc + 2048;                     // 2048 floats (vc @ Wo)

    // 1) vc[h*128+k] = sum_d vp[d] * Wv[h,d,k]   (grid: 8 col-tiles x 16 heads)
    gemv_wmma_f32<128, 256><<<dim3(128 / 16, 16), 256, 0, stream>>>(
        vp, Wv, vc, /*strideB=*/(long long)2048 * 128, /*strideOut=*/128);

    // 2) r = vc @ Wo                             (grid: 128 col-tiles)
    gemv_wmma_f32<2048, 256><<<dim3(2048 / 16, 1), 256, 0, stream>>>(
        vc, Wo, r, /*strideB=*/0, /*strideOut=*/0);

    // 3) broadcast r to all 8192 output rows (64 MiB of global_store_b128)
    broadcast_row_f32<<<dim3(512 / 256, 8192), 256, 0, stream>>>(r, out, 512);
}